// DecoderOutputLayer_68891275427867
// MI455X (gfx1250) — compile-verified
//
#include <hip/hip_runtime.h>
#include <hip/hip_bf16.h>
#include <math.h>

#define BATCH   32
#define DIMQ    512
#define KDIM    1024
#define MZ      2048
#define VOCAB   50257

// GEMM tiling
#define WROWS   128          // vocab rows per block (8 waves x 16)
#define KC      32           // k-chunk per TDM transfer
#define WSTRIDE 34           // padded LDS row stride (dwords): 32 data + 2 pad

typedef __attribute__((ext_vector_type(2))) float v2f;
typedef __attribute__((ext_vector_type(8))) float v8f;
typedef __attribute__((ext_vector_type(4))) unsigned int v4u;
typedef __attribute__((ext_vector_type(4))) int v4i;
typedef __attribute__((ext_vector_type(8))) int v8i;

// ---------------------------------------------------------------------------
// 1) Attention scores: w[b,j] = embsumm[b].memembsumm[b,j] + enc[b].memencs[b,j]
//    + log(memmask[b,j]).  grid = (8 j-chunks, 32 batch), 256 thr (8 waves).
// ---------------------------------------------------------------------------
__global__ __launch_bounds__(256) void scores_kernel(
    const float* __restrict__ enc, const float* __restrict__ embsumm,
    const float* __restrict__ memencs, const float* __restrict__ memembsumm,
    const float* __restrict__ memmask, float* __restrict__ wbuf)
{
    const int b  = blockIdx.y;
    const int jc = blockIdx.x;
    __shared__ float s_q[KDIM];
    for (int i = threadIdx.x; i < DIMQ; i += 256) {
        s_q[i]        = embsumm[b * DIMQ + i];
        s_q[DIMQ + i] = enc[b * DIMQ + i];
    }
    __syncthreads();

    const int lane = threadIdx.x & 31;
    const int wave = threadIdx.x >> 5;
    for (int jj = wave; jj < 256; jj += 8) {
        const int j = jc * 256 + jj;
        const float* ke = memembsumm + ((size_t)b * MZ + j) * DIMQ;
        const float* kc = memencs    + ((size_t)b * MZ + j) * DIMQ;
        float acc = 0.f;
        for (int k = lane * 4; k < DIMQ; k += 128) {
            float4 e  = *(const float4*)(ke + k);
            float4 c  = *(const float4*)(kc + k);
            float4 q0 = *(const float4*)(s_q + k);
            float4 q1 = *(const float4*)(s_q + DIMQ + k);
            acc += e.x * q0.x + e.y * q0.y + e.z * q0.z + e.w * q0.w;
            acc += c.x * q1.x + c.y * q1.y + c.z * q1.z + c.w * q1.w;
        }
        for (int off = 16; off > 0; off >>= 1) acc += __shfl_down(acc, off, 32);
        if (lane == 0)
            wbuf[(size_t)b * MZ + j] = acc + logf(memmask[(size_t)b * MZ + j]);
    }
}

// ---------------------------------------------------------------------------
// 2) Softmax over the 2048 memory slots (per batch row, in place).
// ---------------------------------------------------------------------------
__global__ __launch_bounds__(256) void softmax2048_kernel(float* __restrict__ wa)
{
    const int b = blockIdx.x;
    __shared__ float s_w[MZ];
    __shared__ float s_red[256];
    float* row = wa + (size_t)b * MZ;

    float m = -INFINITY;
    for (int j = threadIdx.x; j < MZ; j += 256) { float v = row[j]; s_w[j] = v; m = fmaxf(m, v); }
    s_red[threadIdx.x] = m; __syncthreads();
    for (int s = 128; s > 0; s >>= 1) {
        if (threadIdx.x < s) s_red[threadIdx.x] = fmaxf(s_red[threadIdx.x], s_red[threadIdx.x + s]);
        __syncthreads();
    }
    const float wmax = s_red[0]; __syncthreads();

    float sum = 0.f;
    for (int j = threadIdx.x; j < MZ; j += 256) { float e = __expf(s_w[j] - wmax); s_w[j] = e; sum += e; }
    s_red[threadIdx.x] = sum; __syncthreads();
    for (int s = 128; s > 0; s >>= 1) {
        if (threadIdx.x < s) s_red[threadIdx.x] += s_red[threadIdx.x + s];
        __syncthreads();
    }
    const float inv = 1.0f / s_red[0];
    for (int j = threadIdx.x; j < MZ; j += 256) row[j] = s_w[j] * inv;
}

// ---------------------------------------------------------------------------
// 3) Partial weighted memory sum (deterministic two-stage reduction).
//    grid = (8 j-chunks, 32 batch).  partial[(b*8+jc)*512 + d]
// ---------------------------------------------------------------------------
__global__ __launch_bounds__(256) void memsum_partial_kernel(
    const float* __restrict__ memencs, const float* __restrict__ alphas,
    float* __restrict__ partial)
{
    const int b = blockIdx.y, jc = blockIdx.x;
    __shared__ float s_a[256];
    const int j0 = jc * 256;
    s_a[threadIdx.x] = alphas[(size_t)b * MZ + j0 + threadIdx.x];
    __syncthreads();

    const int d0 = threadIdx.x, d1 = threadIdx.x + 256;
    const float* base = memencs + ((size_t)b * MZ + j0) * DIMQ;
    float acc0 = 0.f, acc1 = 0.f;
    for (int j = 0; j < 256; ++j) {
        const float a = s_a[j];
        acc0 += a * base[(size_t)j * DIMQ + d0];
        acc1 += a * base[(size_t)j * DIMQ + d1];
    }
    float* p = partial + ((size_t)b * 8 + jc) * DIMQ;
    p[d0] = acc0; p[d1] = acc1;
}

// ---------------------------------------------------------------------------
// 4) logits = H @ W_out^T + b_out + log(unktok),  H = [encsumm, enc] (32x1024)
//    WMMA f32 16x16x4.  Full blocks use the Tensor Data Mover to DMA
//    128x32 W_out tiles into LDS (double buffered, TENSORcnt-tracked,
//    pad_amount inserts 2 dwords per 32 -> bank-conflict-free row stride 34).
//    The ragged last block falls back to clamped per-lane global loads.
// ---------------------------------------------------------------------------
__global__ __launch_bounds__(256) void gemm_logits_kernel(
    const float* __restrict__ enc, const float* __restrict__ encsumm,
    const float* __restrict__ W_out, const float* __restrict__ b_out,
    const float* __restrict__ unktok, float* __restrict__ logits)
{
    __shared__ float s_w[2][WROWS * WSTRIDE];

    const int lane  = threadIdx.x & 31;
    const int wave  = threadIdx.x >> 5;
    const int n     = lane & 15;         // N (vocab) / M (batch) index within tile
    const int khalf = (lane >> 4) * 2;   // 0 or 2: which K pair this lane holds
    const int v0blk = blockIdx.x * WROWS;
    const int v     = v0blk + wave * 16 + n;

    v8f acc0 = {0.f,0.f,0.f,0.f,0.f,0.f,0.f,0.f};
    v8f acc1 = {0.f,0.f,0.f,0.f,0.f,0.f,0.f,0.f};

    if (v0blk + WROWS <= VOCAB) {
        // ---------------- TDM path (full 128-row tile) ----------------
        const unsigned long long gbase =
            (unsigned long long)(uintptr_t)W_out + (unsigned long long)v0blk * (KDIM * 4);
        const unsigned int lds0 = (unsigned int)(uintptr_t)&s_w[0][0];
        const unsigned int lds1 = (unsigned int)(uintptr_t)&s_w[1][0];

        // D# group1: data_size=4B, pad_enable, pad_interval=32dw(code4),
        // pad_amount=2dw(code1); tensor_dim0=1024, tensor_dim1=128,
        // tile_dim0=KC, tile_dim1=WROWS, tensor_dim0_stride=1024.
        const v8i g1 = { (int)0x03120000u,
                         (int)(1024u << 16),
                         (int)(((unsigned)WROWS) << 16),
                         (int)(((unsigned)KC) << 16),
                         (int)WROWS,
                         (int)KDIM,
                         0, 0 };
        const v4i g2 = { 1, 0, 0, 0 };   // tensor_dim2=1, dims 3/4 unused
        const v4i g3 = { 0, 0, 0, 0 };
        const v8i gx = { 0, 0, 0, 0, 0, 0, 0, 0 };  // extra operand (clang-23 6-arg form)

        auto issue = [&](int chunk, int buf) {
            const unsigned long long ga = gbase + (unsigned long long)chunk * (KC * 4);
            const v4u g0 = { 1u,                                   // count=1 (valid D#)
                             (buf ? lds1 : lds0),                  // lds_addr
                             (unsigned int)(ga & 0xFFFFFFFFull),   // global_addr[31:0]
                             (unsigned int)((ga >> 32) & 0x1FFFFFFull) | (2u << 30) }; // [56:32] | type=2
            __builtin_amdgcn_tensor_load_to_lds(g0, g1, g2, g3, gx, 0);
        };

        if (wave == 0) issue(0, 0);

        const int NC = KDIM / KC;        // 32 chunks
        for (int c = 0; c < NC; ++c) {
            const int buf = c & 1;
            if (wave == 0) {
                if (c + 1 < NC) {
                    issue(c + 1, buf ^ 1);                 // prefetch next chunk
                    __builtin_amdgcn_s_wait_tensorcnt(1);  // chunk c landed
                } else {
                    __builtin_amdgcn_s_wait_tensorcnt(0);
                }
            }
            __syncthreads();             // publish tile to all waves

            const float* wt = &s_w[buf][(wave * 16 + n) * WSTRIDE];
            const int kbase = c * KC;
#pragma unroll
            for (int kl = 0; kl < KC; kl += 4) {
                const int kk = kbase + kl + khalf;
                const float* h0 = (kk < DIMQ) ? (encsumm + n * DIMQ + kk)
                                              : (enc + n * DIMQ + (kk - DIMQ));
                const float* h1 = (kk < DIMQ) ? (encsumm + (n + 16) * DIMQ + kk)
                                              : (enc + (n + 16) * DIMQ + (kk - DIMQ));
                const v2f a0 = *(const v2f*)h0;
                const v2f a1 = *(const v2f*)h1;
                const v2f bb = *(const v2f*)(wt + kl + khalf);
                acc0 = __builtin_amdgcn_wmma_f32_16x16x4_f32(false, a0, false, bb,
                                                             (short)0, acc0, false, false);
                acc1 = __builtin_amdgcn_wmma_f32_16x16x4_f32(false, a1, false, bb,
                                                             (short)0, acc1, false, false);
            }
            __syncthreads();             // buffer free before re-DMA
        }
    } else {
        // ---------------- tail path (ragged block, clamped loads) ------------
        const int vc = (v < VOCAB) ? v : (VOCAB - 1);
        const float* wrow = W_out + (size_t)vc * KDIM;
#pragma unroll 4
        for (int k0 = 0; k0 < KDIM; k0 += 4) {
            const int kk = k0 + khalf;
            const float* h0 = (kk < DIMQ) ? (encsumm + n * DIMQ + kk)
                                          : (enc + n * DIMQ + (kk - DIMQ));
            const float* h1 = (kk < DIMQ) ? (encsumm + (n + 16) * DIMQ + kk)
                                          : (enc + (n + 16) * DIMQ + (kk - DIMQ));
            const v2f a0 = *(const v2f*)h0;
            const v2f a1 = *(const v2f*)h1;
            const v2f bb = *(const v2f*)(wrow + kk);
            acc0 = __builtin_amdgcn_wmma_f32_16x16x4_f32(false, a0, false, bb,
                                                         (short)0, acc0, false, false);
            acc1 = __builtin_amdgcn_wmma_f32_16x16x4_f32(false, a1, false, bb,
                                                         (short)0, acc1, false, false);
        }
    }

    if (v < VOCAB) {
        const float bias = b_out[v] + logf(unktok[v]);
        const int mb = (lane >> 4) * 8;   // D layout: lanes 16-31 hold M = 8..15
#pragma unroll
        for (int r = 0; r < 8; ++r) {
            logits[(size_t)(mb + r) * VOCAB + v]      = acc0[r] + bias;
            logits[(size_t)(16 + mb + r) * VOCAB + v] = acc1[r] + bias;
        }
    }
}

// ---------------------------------------------------------------------------
// 5) Per-batch: finalize mem_enc_summ, mixer softmax, vocab softmax,
//    write mix0 * probs_gen into d_out.
// ---------------------------------------------------------------------------
__global__ __launch_bounds__(256) void combine_kernel(
    const float* __restrict__ enc, const float* __restrict__ encsumm,
    const float* __restrict__ W_mix, const float* __restrict__ b_mix,
    const float* __restrict__ logits, const float* __restrict__ partial,
    float* __restrict__ out, float* __restrict__ ws_mix)
{
    const int b = blockIdx.x;
    __shared__ float s_r0[256], s_r1[256];

    // mem_enc_summ (output #2) = sum of 8 deterministic partials
    for (int d = threadIdx.x; d < DIMQ; d += 256) {
        float acc = 0.f;
        for (int jc = 0; jc < 8; ++jc) acc += partial[((size_t)b * 8 + jc) * DIMQ + d];
        out[(size_t)BATCH * VOCAB + (size_t)b * DIMQ + d] = acc;
    }

    // mixer: softmax(h @ W_mix^T + b_mix)
    float p0 = 0.f, p1 = 0.f;
    for (int k = threadIdx.x; k < KDIM; k += 256) {
        const float hv = (k < DIMQ) ? encsumm[b * DIMQ + k] : enc[b * DIMQ + (k - DIMQ)];
        p0 += hv * W_mix[k];
        p1 += hv * W_mix[KDIM + k];
    }
    s_r0[threadIdx.x] = p0; s_r1[threadIdx.x] = p1; __syncthreads();
    for (int s = 128; s > 0; s >>= 1) {
        if (threadIdx.x < s) { s_r0[threadIdx.x] += s_r0[threadIdx.x + s];
                               s_r1[threadIdx.x] += s_r1[threadIdx.x + s]; }
        __syncthreads();
    }
    const float l0 = s_r0[0] + b_mix[0], l1 = s_r1[0] + b_mix[1];
    const float mm = fmaxf(l0, l1);
    const float e0 = __expf(l0 - mm), e1 = __expf(l1 - mm);
    const float mix0 = e0 / (e0 + e1), mix1 = e1 / (e0 + e1);
    if (threadIdx.x == 0) { ws_mix[b * 2] = mix0; ws_mix[b * 2 + 1] = mix1; }
    __syncthreads();

    // vocab softmax, scaled by mix0
    const float* lrow = logits + (size_t)b * VOCAB;
    float m = -INFINITY;
    for (int v = threadIdx.x; v < VOCAB; v += 256) m = fmaxf(m, lrow[v]);
    s_r0[threadIdx.x] = m; __syncthreads();
    for (int s = 128; s > 0; s >>= 1) {
        if (threadIdx.x < s) s_r0[threadIdx.x] = fmaxf(s_r0[threadIdx.x], s_r0[threadIdx.x + s]);
        __syncthreads();
    }
    const float lmax = s_r0[0]; __syncthreads();

    float sum = 0.f;
    for (int v = threadIdx.x; v < VOCAB; v += 256) sum += __expf(lrow[v] - lmax);
    s_r0[threadIdx.x] = sum; __syncthreads();
    for (int s = 128; s > 0; s >>= 1) {
        if (threadIdx.x < s) s_r0[threadIdx.x] += s_r0[threadIdx.x + s];
        __syncthreads();
    }
    const float scale = mix0 / s_r0[0];
    float* orow = out + (size_t)b * VOCAB;
    for (int v = threadIdx.x; v < VOCAB; v += 256) orow[v] = __expf(lrow[v] - lmax) * scale;
}

// ---------------------------------------------------------------------------
// 6) Copy-branch scatter: out[b, memids[b,j]] += mix1[b] * alphas[b,j]
// ---------------------------------------------------------------------------
__global__ __launch_bounds__(256) void scatter_kernel(
    const int* __restrict__ memids, const float* __restrict__ alphas,
    const float* __restrict__ ws_mix, float* __restrict__ out)
{
    const int idx = blockIdx.x * 256 + threadIdx.x;
    if (idx >= BATCH * MZ) return;
    const int b = idx >> 11;
    const float a = alphas[idx] * ws_mix[b * 2 + 1];
    const int tok = memids[idx];
    atomicAdd(&out[(size_t)b * VOCAB + tok], a);
}

// ---------------------------------------------------------------------------
extern "C" void kernel_launch(void* const* d_in, const int* in_sizes, int n_in,
                              void* d_out, int out_size, void* d_ws, size_t ws_size,
                              hipStream_t stream)
{
    (void)in_sizes; (void)n_in; (void)out_size; (void)ws_size;

    const float* enc        = (const float*)d_in[0];
    const float* encsumm    = (const float*)d_in[1];
    const float* embsumm    = (const float*)d_in[2];
    const float* memencs    = (const float*)d_in[3];
    /* d_in[4] memencsumm: multiplied by zeros in reference -> unused */
    const float* memembsumm = (const float*)d_in[5];
    const float* memmask    = (const float*)d_in[6];
    const int*   memids     = (const int*)d_in[7];
    const float* W_out      = (const float*)d_in[8];
    const float* b_out      = (const float*)d_in[9];
    const float* W_mix      = (const float*)d_in[10];
    const float* b_mix      = (const float*)d_in[11];
    const float* unktok     = (const float*)d_in[12];
    float* out = (float*)d_out;

    float* ws_alphas  = (float*)d_ws;                          // B*MZ
    float* ws_logits  = ws_alphas + (size_t)BATCH * MZ;        // B*VOCAB
    float* ws_partial = ws_logits + (size_t)BATCH * VOCAB;     // B*8*512
    float* ws_mix     = ws_partial + (size_t)BATCH * 8 * DIMQ; // B*2

    scores_kernel<<<dim3(8, BATCH), 256, 0, stream>>>(
        enc, embsumm, memencs, memembsumm, memmask, ws_alphas);
    softmax2048_kernel<<<BATCH, 256, 0, stream>>>(ws_alphas);
    memsum_partial_kernel<<<dim3(8, BATCH), 256, 0, stream>>>(
        memencs, ws_alphas, ws_partial);

    const int nblocks = (VOCAB + WROWS - 1) / WROWS;   // 393
    gemm_logits_kernel<<<nblocks, 256, 0, stream>>>(
        enc, encsumm, W_out, b_out, unktok, ws_logits);

    combine_kernel<<<BATCH, 256, 0, stream>>>(
        enc, encsumm, W_mix, b_mix, ws_logits, ws_partial, out, ws_mix);
    scatter_kernel<<<(BATCH * MZ + 255) / 256, 256, 0, stream>>>(
        memids, ws_alphas, ws_mix, out);
}